// ScaledDotProductAttention_47029891891312
// MI455X (gfx1250) — compile-verified
//
#include <hip/hip_runtime.h>

typedef __attribute__((ext_vector_type(2))) float v2f;
typedef __attribute__((ext_vector_type(4))) float v4f;
typedef __attribute__((ext_vector_type(8))) float v8f;

#define S_LEN 2048
#define DK    64
#define KPAD  68   // 32x64 K/V tiles padded to 68 floats/row (bank-conflict free)
#define PPAD  18   // 16x16 P scratch padded to 18 floats/row

// Butterfly lane-xor via ds_swizzle (group-of-32 mode: and=0x1F, or=0, xor=MASK).
// Masks 1,2,4,8 stay within each 16-lane half of the wave32.
template <int MASK>
__device__ __forceinline__ float swz_xor(float x) {
  int i = __builtin_amdgcn_ds_swizzle(__builtin_bit_cast(int, x),
                                      (MASK << 10) | 0x1F);
  return __builtin_bit_cast(float, i);
}

__device__ __forceinline__ float rowmax16(float x) {
  x = fmaxf(x, swz_xor<1>(x));
  x = fmaxf(x, swz_xor<2>(x));
  x = fmaxf(x, swz_xor<4>(x));
  x = fmaxf(x, swz_xor<8>(x));
  return x;
}
__device__ __forceinline__ float rowsum16(float x) {
  x += swz_xor<1>(x);
  x += swz_xor<2>(x);
  x += swz_xor<4>(x);
  x += swz_xor<8>(x);
  return x;
}

__global__ __launch_bounds__(128) void flash_attn_f32_wmma(
    const float* __restrict__ q, const float* __restrict__ k,
    const float* __restrict__ v, float* __restrict__ out) {
  __shared__ float ldsK[32 * KPAD];
  __shared__ float ldsV[32 * KPAD];
  __shared__ float ldsP[4 * 16 * PPAD];

  const int tid  = threadIdx.x;
  const int wave = tid >> 5;
  const int lane = tid & 31;
  const int nl   = lane & 15;   // A: row m, B/C/D: column n
  const int g    = lane >> 4;   // half-wave selector (K-pair / row+8)

  const int bh   = blockIdx.y;              // 0..31  (B*H)
  const int qblk = blockIdx.x;              // 0..31  (2048/64)
  const long base = (long)bh * S_LEN * DK;
  const int q0 = qblk * 64 + wave * 16;

  // ---- Q tile -> registers in WMMA-A layout, pre-scaled by 1/sqrt(64) ----
  v2f qa[16];
  {
    const float* qrow = q + base + (long)(q0 + nl) * DK;
#pragma unroll
    for (int s = 0; s < 16; ++s) {
      v2f a;
      a.x = qrow[4 * s + 2 * g]     * 0.125f;
      a.y = qrow[4 * s + 2 * g + 1] * 0.125f;
      qa[s] = a;
    }
  }

  v8f accO[4];
  float rm[8], rs[8];
#pragma unroll
  for (int t = 0; t < 4; ++t) accO[t] = (v8f){0.f,0.f,0.f,0.f,0.f,0.f,0.f,0.f};
#pragma unroll
  for (int r = 0; r < 8; ++r) { rm[r] = -3.4e38f; rs[r] = 0.f; }

  float* pScr = ldsP + wave * 16 * PPAD;

  for (int key0 = 0; key0 < S_LEN; key0 += 32) {
    __syncthreads();  // previous chunk fully consumed
    {
      // 128 threads cooperatively load 32x64 K and V tiles (fp32, b128 ops)
      const int row = tid >> 3;          // 0..15
      const int col = (tid & 7) * 8;     // 0,8,...,56
      const float* kp  = k + base + (long)(key0 + row) * DK + col;
      const float* vp  = v + base + (long)(key0 + row) * DK + col;
      const float* kp2 = kp + 16 * DK;
      const float* vp2 = vp + 16 * DK;
      *(v4f*)&ldsK[row * KPAD + col]            = *(const v4f*)kp;
      *(v4f*)&ldsK[row * KPAD + col + 4]        = *(const v4f*)(kp + 4);
      *(v4f*)&ldsK[(row + 16) * KPAD + col]     = *(const v4f*)kp2;
      *(v4f*)&ldsK[(row + 16) * KPAD + col + 4] = *(const v4f*)(kp2 + 4);
      *(v4f*)&ldsV[row * KPAD + col]            = *(const v4f*)vp;
      *(v4f*)&ldsV[row * KPAD + col + 4]        = *(const v4f*)(vp + 4);
      *(v4f*)&ldsV[(row + 16) * KPAD + col]     = *(const v4f*)vp2;
      *(v4f*)&ldsV[(row + 16) * KPAD + col + 4] = *(const v4f*)(vp2 + 4);
    }
    __syncthreads();

#pragma unroll
    for (int sub = 0; sub < 2; ++sub) {
      const float* Kt = ldsK + sub * 16 * KPAD;

      // ---- scores: S = (Q/sqrt(d)) @ K^T, 16 fp32 WMMAs over d ----
      v8f sc = (v8f){0.f,0.f,0.f,0.f,0.f,0.f,0.f,0.f};
#pragma unroll
      for (int s = 0; s < 16; ++s) {
        v2f b;
        b.x = Kt[nl * KPAD + 4 * s + 2 * g];
        b.y = Kt[nl * KPAD + 4 * s + 2 * g + 1];
        sc = __builtin_amdgcn_wmma_f32_16x16x4_f32(
            false, qa[s], false, b, (short)0, sc, false, false);
      }

      // ---- online softmax (row stats via 16-lane butterflies) ----
#pragma unroll
      for (int r = 0; r < 8; ++r) {
        float mrow = rowmax16(sc[r]);
        float nm   = fmaxf(rm[r], mrow);
        float corr = __expf(rm[r] - nm);
        rm[r] = nm;
        float p    = __expf(sc[r] - nm);
        float srow = rowsum16(p);
        rs[r] = rs[r] * corr + srow;
        pScr[(r + 8 * g) * PPAD + nl] = p;  // C-layout -> LDS
#pragma unroll
        for (int t = 0; t < 4; ++t) accO[t][r] *= corr;
      }

      // ---- O += P @ V : 4 n-tiles x 4 K-steps of fp32 WMMA ----
#pragma unroll
      for (int t = 0; t < 4; ++t) {
        v8f acc = accO[t];
#pragma unroll
        for (int ks = 0; ks < 4; ++ks) {
          v2f a, b;
          a.x = pScr[nl * PPAD + 4 * ks + 2 * g];
          a.y = pScr[nl * PPAD + 4 * ks + 2 * g + 1];
          const int vr = sub * 16 + 4 * ks + 2 * g;
          b.x = ldsV[vr * KPAD + t * 16 + nl];
          b.y = ldsV[(vr + 1) * KPAD + t * 16 + nl];
          acc = __builtin_amdgcn_wmma_f32_16x16x4_f32(
              false, a, false, b, (short)0, acc, false, false);
        }
        accO[t] = acc;
      }
    }
  }

  // ---- epilogue: divide by running sum, store ----
#pragma unroll
  for (int r = 0; r < 8; ++r) {
    const float inv = 1.0f / rs[r];
    float* orow = out + base + (long)(q0 + r + 8 * g) * DK;
#pragma unroll
    for (int t = 0; t < 4; ++t)
      orow[t * 16 + nl] = accO[t][r] * inv;
  }
}

extern "C" void kernel_launch(void* const* d_in, const int* in_sizes, int n_in,
                              void* d_out, int out_size, void* d_ws, size_t ws_size,
                              hipStream_t stream) {
  (void)in_sizes; (void)n_in; (void)out_size; (void)d_ws; (void)ws_size;
  const float* q = (const float*)d_in[0];
  const float* k = (const float*)d_in[1];
  const float* v = (const float*)d_in[2];
  float* out = (float*)d_out;
  dim3 grid(S_LEN / 64, 2 * 16);  // (query blocks, B*H)
  dim3 block(128);
  flash_attn_f32_wmma<<<grid, block, 0, stream>>>(q, k, v, out);
}